// BipartiteMatchingLoss_2834678415885
// MI455X (gfx1250) — compile-verified
//
#include <hip/hip_runtime.h>
#include <math.h>

// Problem constants (from reference): B=8, Q=900, C1=92, T=1600
#define EPSV      1e-6f
#define NB        8
#define NQ        900
#define NC        92          // 92 = 23 * 4  -> exact K coverage with 16x16x4 f32 WMMA
#define NT        1600
#define QTOT      (NB * NQ)   // 7200 = 450 * 16
#define QTILES    (QTOT / 16) // 450
#define TTILES    (NT / 16)   // 100
#define NTILES    (QTILES * TTILES) // 45000
#define KSTEPS    (NC / 4)    // 23
#define INV_PI2_4 0.40528473456935108f  // 4 / pi^2

typedef __attribute__((ext_vector_type(2))) float v2f;
typedef __attribute__((ext_vector_type(8))) float v8f;

__device__ __forceinline__ float wave_max32(float v) {
#pragma unroll
  for (int off = 16; off > 0; off >>= 1) v = fmaxf(v, __shfl_xor(v, off, 32));
  return v;
}
__device__ __forceinline__ float wave_sum32(float v) {
#pragma unroll
  for (int off = 16; off > 0; off >>= 1) v += __shfl_xor(v, off, 32);
  return v;
}

// ---------------------------------------------------------------------------
// Kernel 1: one wave32 per query.
//   negprob[q][c] = -softmax(logits[q])[c]           (7200 x 92, fp32 exact)
//   predD[q]      = {x1,y1,x2,y2, w*h, atan(w/(h+eps))}  (stride 8 floats)
// ---------------------------------------------------------------------------
__global__ void __launch_bounds__(256) prep_pred_kernel(
    const float* __restrict__ logits, const float* __restrict__ pbox,
    float* __restrict__ negprob, float* __restrict__ predD) {
  const int wave = threadIdx.x >> 5;
  const int lane = threadIdx.x & 31;
  const int q = blockIdx.x * (blockDim.x >> 5) + wave;   // 900 blocks * 8 waves = 7200

  const float* row = logits + (size_t)q * NC;
  const float l0 = row[lane];
  const float l1 = row[lane + 32];
  const float l2 = (lane < NC - 64) ? row[lane + 64] : -3.40282e38f;

  float m = fmaxf(fmaxf(l0, l1), l2);
  m = wave_max32(m);

  const float e0 = expf(l0 - m);
  const float e1 = expf(l1 - m);
  const float e2 = (lane < NC - 64) ? expf(l2 - m) : 0.0f;
  float s = e0 + e1 + e2;
  s = wave_sum32(s);
  const float inv = 1.0f / s;

  float* orow = negprob + (size_t)q * NC;
  orow[lane]      = -e0 * inv;
  orow[lane + 32] = -e1 * inv;
  if (lane < NC - 64) orow[lane + 64] = -e2 * inv;

  if (lane == 0) {
    const float x1 = pbox[q * 4 + 0], y1 = pbox[q * 4 + 1];
    const float x2 = pbox[q * 4 + 2], y2 = pbox[q * 4 + 3];
    const float w = x2 - x1, h = y2 - y1;
    float* pd = predD + q * 8;
    pd[0] = x1; pd[1] = y1; pd[2] = x2; pd[3] = y2;
    pd[4] = w * h;
    pd[5] = atanf(w / (h + EPSV));
  }
}

// ---------------------------------------------------------------------------
// Kernel 2: target descriptors {x1,y1,x2,y2, area, atan} (stride 8 floats)
// ---------------------------------------------------------------------------
__global__ void prep_tgt_kernel(const float* __restrict__ tbox,
                                float* __restrict__ tgtD) {
  const int t = blockIdx.x * blockDim.x + threadIdx.x;
  if (t >= NT) return;
  const float x1 = tbox[t * 4 + 0], y1 = tbox[t * 4 + 1];
  const float x2 = tbox[t * 4 + 2], y2 = tbox[t * 4 + 3];
  const float w = x2 - x1, h = y2 - y1;
  float* td = tgtD + t * 8;
  td[0] = x1; td[1] = y1; td[2] = x2; td[3] = y2;
  td[4] = w * h;
  td[5] = atanf(w / (h + EPSV));
}

// ---------------------------------------------------------------------------
// Kernel 3: one wave32 per 16x16 output tile.
//   Class cost via 23x V_WMMA_F32_16X16X4_F32 against an in-register one-hot
//   B matrix (exact fp32 gather-as-matmul), then fused bbox-L1 + CIoU epilogue
//   added into the accumulator; output written exactly once (coalesced).
//
//   All A fragments (and the target row) are preloaded in one load clause so
//   the wave pays a single load-latency stall, then the 23 WMMAs issue
//   back-to-back; the one-hot cndmask VALU is independent and co-executes.
//
// Register layouts (ISA 7.12.2, f32):
//   A 16x4 : lane lo = M row; VGPR0 <-> K=2*hi, VGPR1 <-> K=2*hi+1
//   B 4x16 : lane lo = N col; VGPR0 <-> K=2*hi, VGPR1 <-> K=2*hi+1
//   C 16x16: VGPR j  <-> M = j + 8*hi, N = lo
// ---------------------------------------------------------------------------
__global__ void __launch_bounds__(256) cost_tile_kernel(
    const float* __restrict__ negprob, const float* __restrict__ predD,
    const float* __restrict__ tgtD, const int* __restrict__ ids,
    float* __restrict__ out) {
  const int wave = threadIdx.x >> 5;
  const int lane = threadIdx.x & 31;
  const int lo = lane & 15;
  const int hi = lane >> 4;

  const int tile  = blockIdx.x * 8 + wave;        // 5625 blocks * 8 waves = 45000
  const int tileQ = tile / TTILES;
  const int tileT = tile - tileQ * TTILES;
  const int qBase = tileQ << 4;
  const int tBase = tileT << 4;

  const int idn = ids[tBase + lo];                // column class id for this lane

  // ---- preload phase: A fragments + target descriptor row (one clause) ----
  const float* aptr = negprob + (size_t)(qBase + lo) * NC + (hi << 1);
  v2f afrag[KSTEPS];
#pragma unroll
  for (int s = 0; s < KSTEPS; ++s) {
    afrag[s].x = aptr[s * 4];                     // A[lo][4s + 2*hi]
    afrag[s].y = aptr[s * 4 + 1];                 // A[lo][4s + 2*hi + 1]
  }
  const float* td = tgtD + (size_t)(tBase + lo) * 8;
  const float4 tb = *reinterpret_cast<const float4*>(td);
  const float2 te = *reinterpret_cast<const float2*>(td + 4);

  // ---- MMA phase: 23 back-to-back WMMAs, one-hot B built in registers ----
  v8f acc = {0.f, 0.f, 0.f, 0.f, 0.f, 0.f, 0.f, 0.f};
#pragma unroll
  for (int s = 0; s < KSTEPS; ++s) {
    const int k0 = (s << 2) + (hi << 1);
    v2f b;
    b.x = (idn == k0)     ? 1.0f : 0.0f;          // B[k0][lo]
    b.y = (idn == k0 + 1) ? 1.0f : 0.0f;          // B[k0+1][lo]
    acc = __builtin_amdgcn_wmma_f32_16x16x4_f32(
        false, afrag[s], false, b, (short)0, acc, false, false);
  }

  // ---- epilogue: hoist the 8 pred descriptor rows, then compute + store ----
  float4 pb[8];
  float2 pe[8];
#pragma unroll
  for (int j = 0; j < 8; ++j) {
    const float* pd = predD + (size_t)(qBase + (hi << 3) + j) * 8;
    pb[j] = *reinterpret_cast<const float4*>(pd);
    pe[j] = *reinterpret_cast<const float2*>(pd + 4);
  }

  const float tarea = te.x, tatan = te.y;
  const int t = tBase + lo;

#pragma unroll
  for (int j = 0; j < 8; ++j) {
    const int q = qBase + (hi << 3) + j;
    const float parea = pe[j].x, patan = pe[j].y;

    // L1 bbox cost
    const float l1c = fabsf(pb[j].x - tb.x) + fabsf(pb[j].y - tb.y) +
                      fabsf(pb[j].z - tb.z) + fabsf(pb[j].w - tb.w);

    // IoU
    const float ix1 = fmaxf(pb[j].x, tb.x), iy1 = fmaxf(pb[j].y, tb.y);
    const float ix2 = fminf(pb[j].z, tb.z), iy2 = fminf(pb[j].w, tb.w);
    const float inter = fmaxf(ix2 - ix1, 0.f) * fmaxf(iy2 - iy1, 0.f);
    const float uni = parea + tarea - inter;
    const float iou = inter / (uni + EPSV);

    // distance penalty
    const float cw = fmaxf(pb[j].z, tb.z) - fminf(pb[j].x, tb.x);
    const float ch = fmaxf(pb[j].w, tb.w) - fminf(pb[j].y, tb.y);
    const float diag = cw * cw + ch * ch;
    const float dx = tb.x + tb.z - pb[j].x - pb[j].z;
    const float dy = tb.y + tb.w - pb[j].y - pb[j].w;
    const float dist = (dx * dx + dy * dy) / (diag + EPSV) * 0.25f;

    // aspect-ratio term (arctans precomputed per box)
    const float dv = tatan - patan;
    const float v = INV_PI2_4 * dv * dv;
    const float alpha = v / (1.0f + EPSV - iou + v);
    const float ciou = iou - dist - alpha * v;

    // cost = bbox + class + (-ciou); acc[j] already holds -prob (class cost)
    out[(size_t)q * NT + t] = l1c + acc[j] - ciou;
  }
}

// ---------------------------------------------------------------------------
// Launch: workspace = negprob (7200*92) | predD (7200*8) | tgtD (1600*8)
//         total ~2.93 MB of fp32
// ---------------------------------------------------------------------------
extern "C" void kernel_launch(void* const* d_in, const int* in_sizes, int n_in,
                              void* d_out, int out_size, void* d_ws, size_t ws_size,
                              hipStream_t stream) {
  (void)in_sizes; (void)n_in; (void)out_size; (void)ws_size;

  const float* logits = (const float*)d_in[0];   // (8,900,92) fp32
  const float* pbox   = (const float*)d_in[1];   // (8,900,4)  fp32
  const int*   ids    = (const int*)d_in[2];     // (1600,)    int32
  const float* tbox   = (const float*)d_in[3];   // (1600,4)   fp32
  float* out = (float*)d_out;                    // (8,900,1600) fp32

  float* negprob = (float*)d_ws;                 // 7200*92
  float* predD   = negprob + (size_t)QTOT * NC;  // 7200*8
  float* tgtD    = predD + (size_t)QTOT * 8;     // 1600*8

  prep_pred_kernel<<<NQ, 256, 0, stream>>>(logits, pbox, negprob, predD);
  prep_tgt_kernel<<<(NT + 255) / 256, 256, 0, stream>>>(tbox, tgtD);
  cost_tile_kernel<<<NTILES / 8, 256, 0, stream>>>(negprob, predD, tgtD, ids, out);
}